// FraudNATHybridModel_65481071406102
// MI455X (gfx1250) — compile-verified
//
#include <hip/hip_runtime.h>
#include <math.h>

typedef float v2f __attribute__((ext_vector_type(2)));
typedef float v8f __attribute__((ext_vector_type(8)));

// ---------------- complex helpers ----------------
__device__ inline float2 cmul(float2 a, float2 b) {
    return make_float2(a.x * b.x - a.y * b.y, a.x * b.y + a.y * b.x);
}
__device__ inline float2 cadd(float2 a, float2 b) {
    return make_float2(a.x + b.x, a.y + b.y);
}

// ---------------- CDNA5 async global->LDS helpers ----------------
// GLOBAL_LOAD_ASYNC_TO_LDS_B128, GVS mode: vdst = LDS byte address,
// vaddr = 32-bit byte offset, saddr = 64-bit SGPR base. Tracked on ASYNCcnt.
__device__ inline void async_ld16(void* lds_dst, const float* gbase, unsigned byte_off) {
    unsigned lds = (unsigned)(uintptr_t)lds_dst;   // LDS aperture: low 32 bits = LDS addr
    asm volatile("global_load_async_to_lds_b128 %0, %1, %2"
                 :: "v"(lds), "v"(byte_off), "s"(gbase) : "memory");
}
__device__ inline void wait_async0() {
    asm volatile("s_wait_asynccnt 0x0" ::: "memory");
}

// ---------------- frozen "random layer" gate table ----------------
// type: 0=rx 1=ry 2=rz 3=crx 4=cnot   (deterministic stand-in sequence,
// same gate/wire distribution as the reference's frozen RAND_SPEC)
__device__ __constant__ int RTYPE[50] = {
    1,3,2,4,0,1,3,2,0,4,
    1,2,3,0,1,4,2,0,3,1,
    2,4,0,1,3,2,0,4,1,2,
    3,0,1,4,2,0,3,1,2,4,
    0,1,3,2,0,4,1,2,3,0};
__device__ __constant__ int RW0[50] = {
    0,1,2,0,3,1,2,3,0,1,
    2,1,3,1,3,2,0,2,0,0,
    2,3,3,1,1,3,0,2,2,1,
    3,1,3,0,0,2,2,0,2,1,
    3,1,0,3,0,3,2,1,1,1};
__device__ __constant__ int RW1[50] = {
    0,3,0,2,0,0,0,0,0,3,
    0,0,2,0,0,1,0,0,1,0,
    0,0,0,0,2,0,0,3,0,0,
    0,0,0,1,0,0,1,0,0,0,
    0,0,3,0,0,2,0,0,0,0};

// ======================================================================
// Kernel 1: fused conv1+relu+pool + conv2+relu+pool + spatial mean.
// One block per image; image tile and conv2 weights streamed into LDS
// with async global->LDS (ASYNCcnt), activations never leave LDS.
// ======================================================================
__global__ __launch_bounds__(256) void kcnn(const float* __restrict__ x,
                                            const float* __restrict__ w1,
                                            const float* __restrict__ b1,
                                            const float* __restrict__ w2,
                                            const float* __restrict__ b2,
                                            float* __restrict__ pooled) {
    __shared__ float xs[784];     // 28x28 input          (196 x 16B chunks)
    __shared__ float w2s[1152];   // conv2 weights        (288 x 16B chunks)
    __shared__ float h1[1568];    // 8 x 14 x 14 after conv1+relu+pool
    __shared__ float w1s[72];
    __shared__ float b1s[8];
    __shared__ float b2s[16];
    __shared__ float c2[784];     // 16 x 7 x 7 after conv2+relu+pool

    const int b   = blockIdx.x;
    const int tid = threadIdx.x;

    // ---- async fills (no VGPR staging) ----
    const unsigned xoff = (unsigned)b * 3136u;            // bytes into x
    if (tid < 196) async_ld16(&xs[tid * 4], x, xoff + (unsigned)tid * 16u);
    for (int i = tid; i < 288; i += 256)
        async_ld16(&w2s[i * 4], w2, (unsigned)i * 16u);
    // ---- tiny tables: plain loads ----
    for (int i = tid; i < 72; i += 256) w1s[i] = w1[i];
    if (tid < 8)  b1s[tid] = b1[tid];
    if (tid < 16) b2s[tid] = b2[tid];
    wait_async0();
    __syncthreads();

    // conv1 (SAME, 28x28) + relu + maxpool2 -> h1[c][14][14]
    for (int it = tid; it < 1568; it += 256) {
        int c = it / 196, rem = it % 196, py = rem / 14, px = rem % 14;
        float m = -1e30f;
        for (int dy = 0; dy < 2; ++dy)
            for (int dx = 0; dx < 2; ++dx) {
                int y = 2 * py + dy, xx = 2 * px + dx;
                float s = b1s[c];
                #pragma unroll
                for (int ky = 0; ky < 3; ++ky) {
                    int iy = y + ky - 1;
                    if (iy < 0 || iy >= 28) continue;
                    #pragma unroll
                    for (int kx = 0; kx < 3; ++kx) {
                        int ix = xx + kx - 1;
                        if (ix < 0 || ix >= 28) continue;
                        s += w1s[c * 9 + ky * 3 + kx] * xs[iy * 28 + ix];
                    }
                }
                m = fmaxf(m, s);
            }
        h1[it] = fmaxf(m, 0.f);   // relu(max) == max(relu) (monotone)
    }
    __syncthreads();

    // conv2 (SAME, 14x14) + relu + maxpool2 -> c2[oc][7][7]
    for (int it = tid; it < 784; it += 256) {
        int oc = it / 49, cell = it % 49, py = cell / 7, px = cell % 7;
        float m = -1e30f;
        for (int dy = 0; dy < 2; ++dy)
            for (int dx = 0; dx < 2; ++dx) {
                int y = 2 * py + dy, xx = 2 * px + dx;
                float s = b2s[oc];
                for (int ic = 0; ic < 8; ++ic) {
                    const float* wr = &w2s[oc * 72 + ic * 9];
                    const float* hr = &h1[ic * 196];
                    #pragma unroll
                    for (int ky = 0; ky < 3; ++ky) {
                        int iy = y + ky - 1;
                        if (iy < 0 || iy >= 14) continue;
                        #pragma unroll
                        for (int kx = 0; kx < 3; ++kx) {
                            int ix = xx + kx - 1;
                            if (ix < 0 || ix >= 14) continue;
                            s += wr[ky * 3 + kx] * hr[iy * 14 + ix];
                        }
                    }
                }
                m = fmaxf(m, s);
            }
        c2[it] = fmaxf(m, 0.f);
    }
    __syncthreads();

    if (tid < 16) {
        float s = 0.f;
        for (int i = 0; i < 49; ++i) s += c2[tid * 49 + i];
        pooled[(size_t)b * 16 + tid] = s * (1.f / 49.f);
    }
}

// ======================================================================
// Kernel 2: encoder. Per-wire 2-vector = Ry*Rx*Rz*Ry |0>, outer product
// -> 16 complex amplitudes. A[b][0..15]=Re, A[b][16..31]=Im.
// ======================================================================
__global__ __launch_bounds__(256) void kenc(const float* __restrict__ pooled,
                                            float* __restrict__ A, int B) {
    int b = blockIdx.x * blockDim.x + threadIdx.x;
    if (b >= B) return;
    const float* p = pooled + (size_t)b * 16;
    float2 v[4][2];
    #pragma unroll
    for (int w = 0; w < 4; ++w) {
        float ta = p[w], tb = p[4 + w], tc = p[8 + w], td = p[12 + w];
        float2 u0 = make_float2(cosf(0.5f * ta), 0.f);
        float2 u1 = make_float2(sinf(0.5f * ta), 0.f);
        float cb = cosf(0.5f * tb), sb = sinf(0.5f * tb);
        u0 = cmul(make_float2(cb, -sb), u0);
        u1 = cmul(make_float2(cb,  sb), u1);
        float cc = cosf(0.5f * tc), sc = sinf(0.5f * tc);
        float2 mi = make_float2(0.f, -sc);
        float2 t0 = cadd(make_float2(cc * u0.x, cc * u0.y), cmul(mi, u1));
        float2 t1 = cadd(cmul(mi, u0), make_float2(cc * u1.x, cc * u1.y));
        float cd = cosf(0.5f * td), sd = sinf(0.5f * td);
        v[w][0] = make_float2(cd * t0.x - sd * t1.x, cd * t0.y - sd * t1.y);
        v[w][1] = make_float2(sd * t0.x + cd * t1.x, sd * t0.y + cd * t1.y);
    }
    float* Ab = A + (size_t)b * 32;
    #pragma unroll
    for (int i = 0; i < 16; ++i) {
        float2 amp = cmul(cmul(v[0][(i >> 3) & 1], v[1][(i >> 2) & 1]),
                          cmul(v[2][(i >> 1) & 1], v[3][i & 1]));
        Ab[i]      = amp.x;
        Ab[16 + i] = amp.y;
    }
}

// ======================================================================
// Kernel 3: compose fixed circuit into 16x16 complex unitary (one block),
// then emit real 32x32 W for the complex->real GEMM.
// ======================================================================
__device__ inline void apply1q(float2* U, int tid, int wire,
                               float2 m00, float2 m01, float2 m10, float2 m11) {
    __syncthreads();
    int p = tid >> 4, c = tid & 15;
    int mask = 8 >> wire, low = mask - 1;
    int r0 = ((p & ~low) << 1) | (p & low);
    int r1 = r0 | mask;
    float2 u0 = U[r0 * 16 + c], u1 = U[r1 * 16 + c];
    U[r0 * 16 + c] = cadd(cmul(m00, u0), cmul(m01, u1));
    U[r1 * 16 + c] = cadd(cmul(m10, u0), cmul(m11, u1));
}

__device__ inline void applyctrl(float2* U, int tid, int wc, int wt,
                                 float2 m00, float2 m01, float2 m10, float2 m11) {
    __syncthreads();
    if (tid < 64) {
        int q = tid >> 4, c = tid & 15;
        int mc = 8 >> wc, mt = 8 >> wt;
        int f[2], nf = 0;
        for (int bit = 8; bit >= 1; bit >>= 1)
            if (bit != mc && bit != mt) f[nf++] = bit;
        int r0 = mc | ((q & 1) ? f[0] : 0) | ((q & 2) ? f[1] : 0);
        int r1 = r0 | mt;
        float2 u0 = U[r0 * 16 + c], u1 = U[r1 * 16 + c];
        U[r0 * 16 + c] = cadd(cmul(m00, u0), cmul(m01, u1));
        U[r1 * 16 + c] = cadd(cmul(m10, u0), cmul(m11, u1));
    }
}

__global__ __launch_bounds__(128) void kunitary(const float* __restrict__ rp, int nrp,
                                                const float* __restrict__ gp,
                                                float* __restrict__ W) {
    __shared__ float2 U[256];   // U[row*16+col]
    int tid = threadIdx.x;
    for (int i = tid; i < 256; i += 128)
        U[i] = make_float2(((i >> 4) == (i & 15)) ? 1.f : 0.f, 0.f);

    int p = 0;
    for (int g = 0; g < 50; ++g) {
        int t = RTYPE[g], w0 = RW0[g], w1 = RW1[g];
        if (t == 4) {   // cnot
            applyctrl(U, tid, w0, w1, make_float2(0, 0), make_float2(1, 0),
                      make_float2(1, 0), make_float2(0, 0));
        } else {
            int pi = (nrp > 0) ? (p % nrp) : 0;
            float th = rp[pi];
            ++p;
            float c = cosf(0.5f * th), s = sinf(0.5f * th);
            if (t == 0)       // rx
                apply1q(U, tid, w0, make_float2(c, 0), make_float2(0, -s),
                        make_float2(0, -s), make_float2(c, 0));
            else if (t == 1)  // ry
                apply1q(U, tid, w0, make_float2(c, 0), make_float2(-s, 0),
                        make_float2(s, 0), make_float2(c, 0));
            else if (t == 2)  // rz
                apply1q(U, tid, w0, make_float2(c, -s), make_float2(0, 0),
                        make_float2(0, 0), make_float2(c, s));
            else              // crx
                applyctrl(U, tid, w0, w1, make_float2(c, 0), make_float2(0, -s),
                          make_float2(0, -s), make_float2(c, 0));
        }
    }
    // fixed tail: rx(gp0,0) ry(gp1,1) rz(gp2,3) crx(gp3,0->2) H(3) SX(2) CNOT(3->0)
    { float th = gp[0], c = cosf(.5f * th), s = sinf(.5f * th);
      apply1q(U, tid, 0, make_float2(c, 0), make_float2(0, -s), make_float2(0, -s), make_float2(c, 0)); }
    { float th = gp[1], c = cosf(.5f * th), s = sinf(.5f * th);
      apply1q(U, tid, 1, make_float2(c, 0), make_float2(-s, 0), make_float2(s, 0), make_float2(c, 0)); }
    { float th = gp[2], c = cosf(.5f * th), s = sinf(.5f * th);
      apply1q(U, tid, 3, make_float2(c, -s), make_float2(0, 0), make_float2(0, 0), make_float2(c, s)); }
    { float th = gp[3], c = cosf(.5f * th), s = sinf(.5f * th);
      applyctrl(U, tid, 0, 2, make_float2(c, 0), make_float2(0, -s), make_float2(0, -s), make_float2(c, 0)); }
    { const float r = 0.70710678118f;
      apply1q(U, tid, 3, make_float2(r, 0), make_float2(r, 0), make_float2(r, 0), make_float2(-r, 0)); }
    apply1q(U, tid, 2, make_float2(.5f, .5f), make_float2(.5f, -.5f),
            make_float2(.5f, -.5f), make_float2(.5f, .5f));
    applyctrl(U, tid, 3, 0, make_float2(0, 0), make_float2(1, 0),
              make_float2(1, 0), make_float2(0, 0));
    __syncthreads();

    // W[k][n]: out[b][n] = sum_k A[b][k] * W[k][n]
    for (int idx = tid; idx < 1024; idx += 128) {
        int k = idx >> 5, n = idx & 31;
        float val;
        if (n < 16) val = (k < 16) ?  U[n * 16 + k].x        : -U[n * 16 + (k - 16)].y;
        else        val = (k < 16) ?  U[(n - 16) * 16 + k].y :  U[(n - 16) * 16 + (k - 16)].x;
        W[idx] = val;
    }
}

// ======================================================================
// Kernel 4: WMMA GEMM  O[B,32] = A[B,32] x W[32,32]  (fp32 WMMA, K=4 steps)
// Per wave: one 16-row tile, two 16-col tiles, 8 k-steps -> 16 v_wmma.
// ======================================================================
__global__ __launch_bounds__(256) void kwmma(const float* __restrict__ A,
                                             const float* __restrict__ W,
                                             float* __restrict__ O, int B) {
    int lane = threadIdx.x & 31;
    int wv   = threadIdx.x >> 5;
    int base = (blockIdx.x * 8 + wv) * 16;
    if (base >= B) return;                 // whole-wave uniform
    int hl = lane >> 4;                    // 0: K pair {0,1}; 1: K pair {2,3}
    int ln = lane & 15;

    v8f acc0 = {0.f, 0.f, 0.f, 0.f, 0.f, 0.f, 0.f, 0.f};
    v8f acc1 = {0.f, 0.f, 0.f, 0.f, 0.f, 0.f, 0.f, 0.f};
    const float* Arow = A + (size_t)(base + ln) * 32;

    #pragma unroll
    for (int kk = 0; kk < 8; ++kk) {
        int kb = kk * 4 + hl * 2;
        v2f a;  a[0]  = Arow[kb];               a[1]  = Arow[kb + 1];
        v2f b0; b0[0] = W[kb * 32 + ln];        b0[1] = W[(kb + 1) * 32 + ln];
        v2f b1; b1[0] = W[kb * 32 + 16 + ln];   b1[1] = W[(kb + 1) * 32 + 16 + ln];
        acc0 = __builtin_amdgcn_wmma_f32_16x16x4_f32(false, a, false, b0,
                                                     (short)0, acc0, false, false);
        acc1 = __builtin_amdgcn_wmma_f32_16x16x4_f32(false, a, false, b1,
                                                     (short)0, acc1, false, false);
    }
    #pragma unroll
    for (int r = 0; r < 8; ++r) {
        int m = r + hl * 8;
        O[(size_t)(base + m) * 32 + ln]      = acc0[r];
        O[(size_t)(base + m) * 32 + 16 + ln] = acc1[r];
    }
}

// ======================================================================
// Kernel 5: PauliZ expectations + per-block partial sums for batchnorm.
// ======================================================================
__global__ __launch_bounds__(256) void kexp(const float* __restrict__ O,
                                            float* __restrict__ exps,
                                            float* __restrict__ partial, int B) {
    __shared__ float red[256];
    int tid = threadIdx.x;
    int b   = blockIdx.x * 256 + tid;
    float e[4] = {0.f, 0.f, 0.f, 0.f};
    if (b < B) {
        const float* ob = O + (size_t)b * 32;
        #pragma unroll
        for (int i = 0; i < 16; ++i) {
            float re = ob[i], im = ob[16 + i];
            float pr = re * re + im * im;
            #pragma unroll
            for (int w = 0; w < 4; ++w)
                e[w] += ((i >> (3 - w)) & 1) ? -pr : pr;
        }
        #pragma unroll
        for (int w = 0; w < 4; ++w) exps[(size_t)b * 4 + w] = e[w];
    }
    for (int s = 0; s < 8; ++s) {
        float v = (b < B) ? ((s < 4) ? e[s] : e[s - 4] * e[s - 4]) : 0.f;
        red[tid] = v;
        __syncthreads();
        for (int off = 128; off > 0; off >>= 1) {
            if (tid < off) red[tid] += red[tid + off];
            __syncthreads();
        }
        if (tid == 0) partial[blockIdx.x * 8 + s] = red[0];
        __syncthreads();
    }
}

// ======================================================================
// Kernel 6: finalize batch stats, fold batchnorm + linear into 5 coeffs.
// ======================================================================
__global__ __launch_bounds__(64) void kstats(const float* __restrict__ partial, int nblk, int B,
                                             const float* __restrict__ gamma,
                                             const float* __restrict__ beta,
                                             const float* __restrict__ fcw,
                                             const float* __restrict__ fcb,
                                             float* __restrict__ coef) {
    __shared__ float sums[8];
    __shared__ float terms[4];
    int tid = threadIdx.x;
    if (tid < 8) {
        float s = 0.f;
        for (int i = 0; i < nblk; ++i) s += partial[i * 8 + tid];
        sums[tid] = s;
    }
    __syncthreads();
    if (tid < 4) {
        float mu  = sums[tid] / (float)B;
        float var = sums[4 + tid] / (float)B - mu * mu;
        float inv = 1.f / sqrtf(var + 1e-5f);
        coef[tid]  = fcw[tid] * gamma[tid] * inv;
        terms[tid] = fcw[tid] * (beta[tid] - gamma[tid] * mu * inv);
    }
    __syncthreads();
    if (tid == 0) coef[4] = fcb[0] + terms[0] + terms[1] + terms[2] + terms[3];
}

// ======================================================================
// Kernel 7: final output  out[b] = bias + sum_w s_w * e_w
// ======================================================================
__global__ __launch_bounds__(256) void kout(const float* __restrict__ exps,
                                            const float* __restrict__ coef,
                                            float* __restrict__ out, int B) {
    int b = blockIdx.x * blockDim.x + threadIdx.x;
    if (b >= B) return;
    out[b] = coef[4] + coef[0] * exps[(size_t)b * 4 + 0]
                     + coef[1] * exps[(size_t)b * 4 + 1]
                     + coef[2] * exps[(size_t)b * 4 + 2]
                     + coef[3] * exps[(size_t)b * 4 + 3];
}

// ======================================================================
extern "C" void kernel_launch(void* const* d_in, const int* in_sizes, int n_in,
                              void* d_out, int out_size, void* d_ws, size_t ws_size,
                              hipStream_t stream) {
    const float* x       = (const float*)d_in[0];
    const float* conv1_w = (const float*)d_in[1];
    const float* conv1_b = (const float*)d_in[2];
    const float* conv2_w = (const float*)d_in[3];
    const float* conv2_b = (const float*)d_in[4];
    const float* gate_p  = (const float*)d_in[5];
    const float* rand_p  = (const float*)d_in[6];
    const float* bn_g    = (const float*)d_in[7];
    const float* bn_b    = (const float*)d_in[8];
    const float* fc_w    = (const float*)d_in[9];
    const float* fc_b    = (const float*)d_in[10];
    float* out = (float*)d_out;

    const int B    = in_sizes[0] / 784;
    const int nrp  = in_sizes[6];
    const int nblk = (B + 255) / 256;

    float* ws      = (float*)d_ws;
    float* W       = ws;                  // 1024 floats
    float* coef    = ws + 1024;           // 16
    float* partial = ws + 1040;           // nblk*8 (<= 1008)
    float* pooled  = ws + 2048;           // B*16
    float* A       = pooled + (size_t)B * 16;   // B*32
    float* O       = A + (size_t)B * 32;        // B*32
    float* exps    = O + (size_t)B * 32;        // B*4

    kcnn<<<B, 256, 0, stream>>>(x, conv1_w, conv1_b, conv2_w, conv2_b, pooled);
    kunitary<<<1, 128, 0, stream>>>(rand_p, nrp, gate_p, W);
    kenc<<<(B + 255) / 256, 256, 0, stream>>>(pooled, A, B);
    kwmma<<<(B + 127) / 128, 256, 0, stream>>>(A, W, O, B);
    kexp<<<nblk, 256, 0, stream>>>(O, exps, partial, B);
    kstats<<<1, 64, 0, stream>>>(partial, nblk, B, bn_g, bn_b, fc_w, fc_b, coef);
    kout<<<(B + 255) / 256, 256, 0, stream>>>(exps, coef, out, B);
}